// SS2D_51857435132176
// MI455X (gfx1250) — compile-verified
//
#include <hip/hip_runtime.h>
#include <math.h>

#define D_MODEL 192
#define D_STATE 16
#define D_INNER 384
#define DT_RANK 12
#define NB      4
#define HH      32
#define WW      32
#define LTOK    (HH * WW)      // 1024 tokens per batch
#define NTOK    (NB * LTOK)    // 4096 tokens total
#define XDBL_LD 48             // 44 cols (12 dt + 16 B + 16 C) padded to 48

typedef float v2f __attribute__((ext_vector_type(2)));
typedef float v8f __attribute__((ext_vector_type(8)));
typedef int   i32x4 __attribute__((vector_size(16)));

// ---------------------------------------------------------------------------
// Async global -> LDS copies (CDNA5 GLOBAL_LOAD_ASYNC_TO_LDS_*, ASYNCcnt).
// Builtin signatures (from hipcc diagnostics):
//   b128: (i32x4 AS(1)*, i32x4 AS(3)*, imm int offset, imm int cpol)
//   b32 : (int   AS(1)*, int   AS(3)*, imm int offset, imm int cpol)
// ---------------------------------------------------------------------------
#if defined(__has_builtin)
#  if __has_builtin(__builtin_amdgcn_global_load_async_to_lds_b128)
#    define HAVE_ASYNC_LDS 1
#  endif
#endif

__device__ __forceinline__ void g2l_b128(const float* __restrict__ g, float* l) {
#ifdef HAVE_ASYNC_LDS
  __builtin_amdgcn_global_load_async_to_lds_b128(
      (__attribute__((address_space(1))) i32x4*)
          (__attribute__((address_space(1))) const void*)g,
      (__attribute__((address_space(3))) i32x4*)
          (__attribute__((address_space(3))) void*)l,
      0, 0);
#else
  *(float4*)l = *(const float4*)g;
#endif
}

__device__ __forceinline__ void g2l_b32(const float* __restrict__ g, float* l) {
#ifdef HAVE_ASYNC_LDS
  __builtin_amdgcn_global_load_async_to_lds_b32(
      (__attribute__((address_space(1))) int*)
          (__attribute__((address_space(1))) const void*)g,
      (__attribute__((address_space(3))) int*)
          (__attribute__((address_space(3))) void*)l,
      0, 0);
#else
  *l = *g;
#endif
}

__device__ __forceinline__ void wait_async_and_barrier() {
#ifdef HAVE_ASYNC_LDS
#  if __has_builtin(__builtin_amdgcn_s_wait_asynccnt)
  __builtin_amdgcn_s_wait_asynccnt(0);
#  endif
#endif
  __syncthreads();
}

// ---------------------------------------------------------------------------
// Stage a K x 16 panel of B (row-major, leading dim ldb), columns
// [col0, col0+16), into bsm[K*16].  Columns >= nvalid are zero-filled
// (ragged W_x case).  Fast path: 4 async b128 per row.
// ---------------------------------------------------------------------------
__device__ __forceinline__ void stage_panel(const float* __restrict__ B, int ldb,
                                            int nvalid, int col0, int K,
                                            float* bsm) {
  const int tid = threadIdx.x;
  if (col0 + 16 <= nvalid) {
    const int nvec = K * 4;                         // 4 x 16B per row
    for (int i = tid; i < nvec; i += (int)blockDim.x) {
      const int k  = i >> 2;
      const int c4 = (i & 3) * 4;
      g2l_b128(B + k * ldb + col0 + c4, bsm + k * 16 + c4);
    }
  } else {
    for (int i = tid; i < K * 16; i += (int)blockDim.x) {
      const int k = i >> 4;
      const int c = i & 15;
      if (col0 + c < nvalid) g2l_b32(B + k * ldb + col0 + c, bsm + i);
      else                   bsm[i] = 0.f;
    }
  }
  wait_async_and_barrier();
}

// ---------------------------------------------------------------------------
// One 16x16 f32 tile via V_WMMA_F32_16X16X4_F32, A from global (row-major),
// B panel from LDS (bsm[k*16 + c]).  ISA layouts per CDNA5 ISA:
//   A 16x4 : lanes 0-15 K={0,1}, lanes 16-31 K={2,3}; M = lane&15
//   B 4x16 : mirrored; N = lane&15
//   C/D    : VGPR r -> rows r / r+8 across lane halves; N = lane&15
// ---------------------------------------------------------------------------
__device__ __forceinline__ v8f wmma_tile_lds(const float* __restrict__ A, int lda,
                                             const float* __restrict__ bsm,
                                             int row0, int K) {
  v8f acc = {0.f, 0.f, 0.f, 0.f, 0.f, 0.f, 0.f, 0.f};
  const int lane = threadIdx.x & 31;
  const int half = lane >> 4;
  const int r    = lane & 15;
  for (int k0 = 0; k0 < K; k0 += 4) {
    const int ka = k0 + half * 2;
    v2f a = *(const v2f*)(A + (row0 + r) * lda + ka);   // 8B-aligned pair
    v2f b;
    b.x = bsm[ka * 16 + r];
    b.y = bsm[ka * 16 + 16 + r];
    acc = __builtin_amdgcn_wmma_f32_16x16x4_f32(
        /*neg_a=*/false, a, /*neg_b=*/false, b,
        /*c_mod=*/(short)0, acc, /*reuse_a=*/false, /*reuse_b=*/false);
  }
  return acc;
}

// Block layout for all GEMMs: blockIdx.x = rowGroup * NCT + colTile.
// 256 threads = 8 waves; wave w computes row tile (rowGroup*8 + w); all 8
// waves share one LDS-staged 16-column weight panel.

// ---------------------------------------------------------------------------
// K1: xz = x @ W_in ; split columns into x_val (0..383) and z (384..767)
// ---------------------------------------------------------------------------
__global__ void k_inproj(const float* __restrict__ x, const float* __restrict__ W_in,
                         float* __restrict__ xval, float* __restrict__ z) {
  __shared__ __align__(16) float bsm[D_MODEL * 16];          // 12 KB
  const int NCT  = (2 * D_INNER) / 16;                       // 48
  const int ct   = blockIdx.x % NCT;
  const int rg   = blockIdx.x / NCT;
  const int col0 = ct * 16;
  stage_panel(W_in, 2 * D_INNER, 2 * D_INNER, col0, D_MODEL, bsm);

  const int wave = threadIdx.x >> 5;
  const int row0 = (rg * 8 + wave) * 16;
  v8f acc = wmma_tile_lds(x, D_MODEL, bsm, row0, D_MODEL);

  const int lane = threadIdx.x & 31;
  const int half = lane >> 4;
  const int r    = lane & 15;
  #pragma unroll
  for (int i = 0; i < 8; i++) {
    const int rr = row0 + i + 8 * half;
    const int cc = col0 + r;
    if (cc < D_INNER) xval[rr * D_INNER + cc] = acc[i];
    else              z[rr * D_INNER + (cc - D_INNER)] = acc[i];
  }
}

// ---------------------------------------------------------------------------
// K2: depthwise 3x3 SAME conv + bias + SiLU  ->  u
// ---------------------------------------------------------------------------
__global__ void k_conv_silu(const float* __restrict__ xval, const float* __restrict__ cw,
                            const float* __restrict__ cb, float* __restrict__ u) {
  const int idx = blockIdx.x * blockDim.x + threadIdx.x;     // t*384 + d
  if (idx >= NTOK * D_INNER) return;
  const int d = idx % D_INNER;
  const int t = idx / D_INNER;
  const int w = t % WW;
  const int h = (t / WW) % HH;
  const int b = t / LTOK;
  float acc = cb[d];
  #pragma unroll
  for (int dy = 0; dy < 3; dy++) {
    const int hy = h + dy - 1;
    if (hy < 0 || hy >= HH) continue;
    #pragma unroll
    for (int dx = 0; dx < 3; dx++) {
      const int wx = w + dx - 1;
      if (wx < 0 || wx >= WW) continue;
      acc += cw[d * 9 + dy * 3 + dx] * xval[((b * HH + hy) * WW + wx) * D_INNER + d];
    }
  }
  u[idx] = acc / (1.f + __expf(-acc));   // SiLU
}

// ---------------------------------------------------------------------------
// K3a: x_dbl = u @ W_x  -> [dt_raw(12) | B(16) | C(16)] stored with ld 48
// ---------------------------------------------------------------------------
__global__ void k_xdbl(const float* __restrict__ u, const float* __restrict__ W_x,
                       float* __restrict__ xdbl) {
  __shared__ __align__(16) float bsm[D_INNER * 16];          // 24 KB
  const int NCT  = XDBL_LD / 16;                             // 3
  const int ct   = blockIdx.x % NCT;
  const int rg   = blockIdx.x / NCT;
  const int col0 = ct * 16;
  stage_panel(W_x, DT_RANK + 2 * D_STATE, DT_RANK + 2 * D_STATE /*44*/, col0,
              D_INNER, bsm);                                 // pads cols 44..47

  const int wave = threadIdx.x >> 5;
  const int row0 = (rg * 8 + wave) * 16;
  v8f acc = wmma_tile_lds(u, D_INNER, bsm, row0, D_INNER);

  const int lane = threadIdx.x & 31;
  const int half = lane >> 4;
  const int r    = lane & 15;
  #pragma unroll
  for (int i = 0; i < 8; i++) {
    const int rr = row0 + i + 8 * half;
    xdbl[rr * XDBL_LD + col0 + r] = acc[i];                  // pad cols -> zeros
  }
}

// ---------------------------------------------------------------------------
// K3b: dt = softplus(x_dbl[:, :12] @ W_dt + b_dt)   (K = 12)
// ---------------------------------------------------------------------------
__global__ void k_dtproj(const float* __restrict__ xdbl, const float* __restrict__ W_dt,
                         const float* __restrict__ b_dt, float* __restrict__ dt) {
  __shared__ __align__(16) float bsm[DT_RANK * 16];          // 768 B
  const int NCT  = D_INNER / 16;                             // 24
  const int ct   = blockIdx.x % NCT;
  const int rg   = blockIdx.x / NCT;
  const int col0 = ct * 16;
  stage_panel(W_dt, D_INNER, D_INNER, col0, DT_RANK, bsm);

  const int wave = threadIdx.x >> 5;
  const int row0 = (rg * 8 + wave) * 16;
  v8f acc = wmma_tile_lds(xdbl, XDBL_LD, bsm, row0, DT_RANK);

  const int lane = threadIdx.x & 31;
  const int half = lane >> 4;
  const int r    = lane & 15;
  #pragma unroll
  for (int i = 0; i < 8; i++) {
    const int rr = row0 + i + 8 * half;
    const int cc = col0 + r;
    float v = acc[i] + b_dt[cc];
    v = (v > 20.f) ? v : log1pf(__expf(v));                  // softplus
    dt[rr * D_INNER + cc] = v;
  }
}

// ---------------------------------------------------------------------------
// K4: selective scan. One block per (batch, direction); one thread per inner
// channel d. h[16] and A[d,:] live in VGPRs; 1024 sequential steps.
// Directional permutations reduce to the token-index mapping, so all four
// scans atomically accumulate into the same per-token y buffer.
// ---------------------------------------------------------------------------
__global__ void k_scan(const float* __restrict__ u, const float* __restrict__ dt,
                       const float* __restrict__ xdbl, const float* __restrict__ A_log,
                       float* __restrict__ y) {
  const int b   = blockIdx.x & 3;
  const int dir = blockIdx.x >> 2;                // 0..3
  const int d   = threadIdx.x;                    // 0..383
  float a[D_STATE], h[D_STATE];
  #pragma unroll
  for (int n = 0; n < D_STATE; n++) {
    a[n] = -__expf(A_log[d * D_STATE + n]);
    h[n] = 0.f;
  }
  for (int l = 0; l < LTOK; l++) {
    const int p = (dir & 1) ? (LTOK - 1 - l) : l;
    int hh, ww;
    if (dir < 2) { hh = p >> 5; ww = p & 31; }    // row-major order
    else         { ww = p >> 5; hh = p & 31; }    // column-major order
    const int t = b * LTOK + hh * WW + ww;
    const float dtv = dt[t * D_INNER + d];
    const float uv  = u [t * D_INNER + d];
    const float* bc = xdbl + t * XDBL_LD;
    float yv = 0.f;
    #pragma unroll
    for (int n = 0; n < D_STATE; n++) {
      const float dA = __expf(dtv * a[n]);
      h[n] = dA * h[n] + (dtv * bc[DT_RANK + n]) * uv;
      yv  += h[n] * bc[DT_RANK + D_STATE + n];
    }
    atomicAdd(&y[t * D_INNER + d], yv);
  }
}

// ---------------------------------------------------------------------------
// K5a: y += 4*u*D ; LayerNorm over D_INNER ; gate with SiLU(z)  -> g
// ---------------------------------------------------------------------------
__global__ void k_norm_gate(const float* __restrict__ y, const float* __restrict__ u,
                            const float* __restrict__ Dp, const float* __restrict__ gamma,
                            const float* __restrict__ beta, const float* __restrict__ z,
                            float* __restrict__ g) {
  const int t = blockIdx.x;
  const int d = threadIdx.x;                      // 0..383, 12 waves
  __shared__ float red1[12], red2[12];
  __shared__ float smu, srstd;
  float v = y[t * D_INNER + d] + 4.f * u[t * D_INNER + d] * Dp[d];

  float s = v;
  #pragma unroll
  for (int off = 16; off > 0; off >>= 1) s += __shfl_down(s, off, 32);
  if ((d & 31) == 0) red1[d >> 5] = s;
  __syncthreads();
  if (d == 0) {
    float tot = 0.f;
    #pragma unroll
    for (int i = 0; i < 12; i++) tot += red1[i];
    smu = tot / (float)D_INNER;
  }
  __syncthreads();
  const float mu = smu;
  float dv = v - mu;
  float s2 = dv * dv;
  #pragma unroll
  for (int off = 16; off > 0; off >>= 1) s2 += __shfl_down(s2, off, 32);
  if ((d & 31) == 0) red2[d >> 5] = s2;
  __syncthreads();
  if (d == 0) {
    float tot = 0.f;
    #pragma unroll
    for (int i = 0; i < 12; i++) tot += red2[i];
    srstd = rsqrtf(tot / (float)D_INNER + 1e-5f);
  }
  __syncthreads();
  const float zn   = z[t * D_INNER + d];
  const float gate = zn / (1.f + __expf(-zn));    // SiLU(z)
  g[t * D_INNER + d] = (dv * srstd * gamma[d] + beta[d]) * gate;
}

// ---------------------------------------------------------------------------
// K5b: out = g @ W_out  (4096x384 @ 384x192)
// ---------------------------------------------------------------------------
__global__ void k_outproj(const float* __restrict__ g, const float* __restrict__ W_out,
                          float* __restrict__ out) {
  __shared__ __align__(16) float bsm[D_INNER * 16];          // 24 KB
  const int NCT  = D_MODEL / 16;                             // 12
  const int ct   = blockIdx.x % NCT;
  const int rg   = blockIdx.x / NCT;
  const int col0 = ct * 16;
  stage_panel(W_out, D_MODEL, D_MODEL, col0, D_INNER, bsm);

  const int wave = threadIdx.x >> 5;
  const int row0 = (rg * 8 + wave) * 16;
  v8f acc = wmma_tile_lds(g, D_INNER, bsm, row0, D_INNER);

  const int lane = threadIdx.x & 31;
  const int half = lane >> 4;
  const int r    = lane & 15;
  #pragma unroll
  for (int i = 0; i < 8; i++) {
    const int rr = row0 + i + 8 * half;
    out[rr * D_MODEL + col0 + r] = acc[i];
  }
}

// ---------------------------------------------------------------------------
extern "C" void kernel_launch(void* const* d_in, const int* in_sizes, int n_in,
                              void* d_out, int out_size, void* d_ws, size_t ws_size,
                              hipStream_t stream) {
  const float* x      = (const float*)d_in[0];
  const float* W_in   = (const float*)d_in[1];
  const float* conv_w = (const float*)d_in[2];
  const float* conv_b = (const float*)d_in[3];
  const float* W_x    = (const float*)d_in[4];
  const float* W_dt   = (const float*)d_in[5];
  const float* b_dt   = (const float*)d_in[6];
  const float* A_log  = (const float*)d_in[7];
  const float* D_par  = (const float*)d_in[8];
  const float* W_out  = (const float*)d_in[9];
  const float* gamma  = (const float*)d_in[10];
  const float* beta   = (const float*)d_in[11];
  float* out = (float*)d_out;

  float* ws     = (float*)d_ws;
  float* f_xval = ws;                              // NTOK*D_INNER
  float* f_z    = f_xval + (size_t)NTOK * D_INNER; // NTOK*D_INNER
  float* f_u    = f_z    + (size_t)NTOK * D_INNER; // NTOK*D_INNER
  float* f_dt   = f_u    + (size_t)NTOK * D_INNER; // NTOK*D_INNER
  float* f_y    = f_dt   + (size_t)NTOK * D_INNER; // NTOK*D_INNER
  float* f_xdbl = f_y    + (size_t)NTOK * D_INNER; // NTOK*XDBL_LD
  float* f_g    = f_xval;                          // reuse (x_val dead after conv)

  (void)hipMemsetAsync(f_y, 0, (size_t)NTOK * D_INNER * sizeof(float), stream);

  // GEMM grids: (rowGroups = 4096/128 = 32) x (col tiles)
  k_inproj   <<<32 * 48, 256, 0, stream>>>(x, W_in, f_xval, f_z);
  k_conv_silu<<<(NTOK * D_INNER + 255) / 256, 256, 0, stream>>>(f_xval, conv_w, conv_b, f_u);
  k_xdbl     <<<32 * 3,  256, 0, stream>>>(f_u, W_x, f_xdbl);
  k_dtproj   <<<32 * 24, 256, 0, stream>>>(f_xdbl, W_dt, b_dt, f_dt);
  k_scan     <<<16, D_INNER, 0, stream>>>(f_u, f_dt, f_xdbl, A_log, f_y);
  k_norm_gate<<<NTOK, D_INNER, 0, stream>>>(f_y, f_u, D_par, gamma, beta, f_z, f_g);
  k_outproj  <<<32 * 12, 256, 0, stream>>>(f_g, W_out, out);
}